// SparseMoELayer_67370857005586
// MI455X (gfx1250) — compile-verified
//
#include <hip/hip_runtime.h>
#include <math.h>

// Problem constants (fixed by the reference).
#define B_ 16
#define S_ 2048
#define NTOK (B_ * S_)      // 32768 tokens
#define D_ 240
#define E_ 8
#define CAP NTOK            // worst-case tokens routed to one expert
#define XS_STRIDE 244       // padded LDS row stride (multiple of 4, avoids bank conflicts)
#define TILE_TOK 64         // tokens per expert-GEMM block
#define BLOCK2 256          // 8 waves; each wave owns 2 output N-tiles

typedef __attribute__((ext_vector_type(2))) float v2f;
typedef __attribute__((ext_vector_type(8))) float v8f;

// Truncate a generic (flat) pointer to its wave-relative LDS byte offset.
// For HIP __shared__ memory, flat = {SHARED_BASE, 32-bit LDS offset}.
__device__ __forceinline__ unsigned lds_u32(const void* p) {
    return (unsigned)(unsigned long long)(uintptr_t)p;
}

// Async global -> LDS copy of 16 bytes (CDNA5; tracked by ASYNCcnt).
__device__ __forceinline__ void async_copy_b128(unsigned lds_addr, unsigned voff,
                                                const void* base) {
    asm volatile("global_load_async_to_lds_b128 %0, %1, %2"
                 :: "v"(lds_addr), "v"(voff),
                    "s"((unsigned long long)(uintptr_t)base)
                 : "memory");
}

__device__ __forceinline__ void wait_asynccnt0() {
    asm volatile("s_wait_asynccnt 0" ::: "memory");
}

// ---------------------------------------------------------------------------
// Kernel 0: zero output accumulator + per-expert routing counters.
// ---------------------------------------------------------------------------
__global__ void moe_zero_kernel(float* __restrict__ out, int n, int* __restrict__ cnt) {
    int tid = blockIdx.x * blockDim.x + threadIdx.x;
    if (tid < E_) cnt[tid] = 0;
    for (int i = tid; i < n; i += gridDim.x * blockDim.x) out[i] = 0.0f;
}

// ---------------------------------------------------------------------------
// Kernel 1: gating. One wave32 per token: 8 logits via lane-strided dot
// products (coalesced x reads), shuffle reduction, top-2 + renormalize,
// atomic append to per-expert compacted (token, weight) lists.
// ---------------------------------------------------------------------------
__global__ __launch_bounds__(256) void moe_gate_kernel(
    const float* __restrict__ x, const float* __restrict__ gw,
    const float* __restrict__ gb, int* __restrict__ cnt,
    int* __restrict__ tok, float* __restrict__ wt) {
    const int lane = threadIdx.x & 31;
    const int wave = threadIdx.x >> 5;
    const int t = blockIdx.x * 8 + wave;
    if (t >= NTOK) return;

    float acc[E_];
#pragma unroll
    for (int e = 0; e < E_; ++e) acc[e] = 0.0f;

    for (int d = lane; d < D_; d += 32) {
        float xv = x[t * D_ + d];
#pragma unroll
        for (int e = 0; e < E_; ++e) acc[e] += xv * gw[e * D_ + d];
    }
#pragma unroll
    for (int e = 0; e < E_; ++e) {
#pragma unroll
        for (int off = 16; off > 0; off >>= 1)
            acc[e] += __shfl_down(acc[e], off, 32);
    }

    if (lane == 0) {
        float l[E_];
#pragma unroll
        for (int e = 0; e < E_; ++e) l[e] = acc[e] + gb[e];
        // top-1 (first occurrence on ties, matching jax top_k)
        int i0 = 0;
#pragma unroll
        for (int e = 1; e < E_; ++e) if (l[e] > l[i0]) i0 = e;
        // top-2
        int i1 = (i0 == 0) ? 1 : 0;
#pragma unroll
        for (int e = 0; e < E_; ++e) if (e != i0 && l[e] > l[i1]) i1 = e;
        // softmax over all E then renormalize over top-2 == 2-way softmax:
        float w0 = 1.0f / (1.0f + __expf(l[i1] - l[i0]));
        float w1 = 1.0f - w0;
        int p0 = atomicAdd(&cnt[i0], 1);
        tok[i0 * CAP + p0] = t;  wt[i0 * CAP + p0] = w0;
        int p1 = atomicAdd(&cnt[i1], 1);
        tok[i1 * CAP + p1] = t;  wt[i1 * CAP + p1] = w1;
    }
}

// ---------------------------------------------------------------------------
// Kernel 2: per-expert grouped GEMM with V_WMMA_F32_16X16X4_F32.
//   Y[64,240] = Xg[64,240] @ W_e^T, scaled per-row by gate weight,
//   atomically accumulated into out.
// Block: 256 threads = 8 waves; wave w owns N-tiles {w, w+8} (wave 7 computes
// a clamped duplicate of its first tile to keep WMMA control flow uniform).
// Each A-fragment load now feeds 2 WMMAs -> 0.75 LDS b64 loads per WMMA.
// Staging uses GLOBAL_LOAD_ASYNC_TO_LDS_B128 (no VGPR round-trip).
// ---------------------------------------------------------------------------
__global__ __launch_bounds__(BLOCK2) void moe_expert_kernel(
    const float* __restrict__ x, const float* __restrict__ ew,
    const float* __restrict__ eb, const int* __restrict__ cnt,
    const int* __restrict__ tok, const float* __restrict__ wt,
    float* __restrict__ out) {
    extern __shared__ float smem[];
    float* Xs   = smem;                         // TILE_TOK * XS_STRIDE
    float* Ws   = Xs + TILE_TOK * XS_STRIDE;    // D_ * XS_STRIDE
    float* Bs   = Ws + D_ * XS_STRIDE;          // D_
    float* Wts  = Bs + D_;                      // TILE_TOK
    int*   Toks = (int*)(Wts + TILE_TOK);       // TILE_TOK

    const int e    = blockIdx.y;
    const int ne   = cnt[e];
    const int base = blockIdx.x * TILE_TOK;
    if (base >= ne) return;                     // uniform early-exit

    const int tid = threadIdx.x;

    // Stage routing metadata for this tile.
    if (tid < TILE_TOK) {
        int idx = base + tid;
        if (idx < ne) { Toks[tid] = tok[e * CAP + idx]; Wts[tid] = wt[e * CAP + idx]; }
        else          { Toks[tid] = -1;                 Wts[tid] = 0.0f; }
    }
    if (tid < D_ / 4) {
        *(float4*)&Bs[tid * 4] = *(const float4*)&eb[e * D_ + tid * 4];
    }
    __syncthreads();   // Toks needed by the X gather below

    // Async-stage full W_e (230 KB) into LDS: 14400 x b128, L2-resident source.
    const float* We = ew + (size_t)e * D_ * D_;
    const unsigned ws_base = lds_u32(Ws);
    for (int f = tid; f < D_ * (D_ / 4); f += BLOCK2) {
        int r = f / (D_ / 4), c = f % (D_ / 4);
        async_copy_b128(ws_base + (unsigned)((r * XS_STRIDE + c * 4) * 4),
                        (unsigned)((r * D_ + c * 4) * 4), We);
    }
    // Async-gather routed token rows of x into LDS; zero padded rows via DS.
    const unsigned xs_base = lds_u32(Xs);
    for (int f = tid; f < TILE_TOK * (D_ / 4); f += BLOCK2) {
        int r = f / (D_ / 4), c = f % (D_ / 4);
        int tk = Toks[r];
        unsigned dst = xs_base + (unsigned)((r * XS_STRIDE + c * 4) * 4);
        if (tk >= 0) {
            async_copy_b128(dst, (unsigned)((tk * D_ + c * 4) * 4), x);
        } else {
            *(float4*)&Xs[r * XS_STRIDE + c * 4] = make_float4(0.f, 0.f, 0.f, 0.f);
        }
    }
    wait_asynccnt0();  // per-wave async completion, then cross-wave barrier
    __syncthreads();

    const int lane = tid & 31;
    const int wave = tid >> 5;                 // 0..7
    const int lo = lane & 15;
    const int hi = lane >> 4;                  // 0 or 1
    const int n0a = wave * 16;                 // first N-tile
    const bool has2 = (wave < 7);
    const int n0b = has2 ? (wave + 8) * 16 : n0a;  // second N-tile (clamped dup)

    // 8 independent accumulator chains: 4 M-tiles x 2 N-tiles.
    v8f acc[4][2];
    const v8f vzero = {0.f, 0.f, 0.f, 0.f, 0.f, 0.f, 0.f, 0.f};
#pragma unroll
    for (int m = 0; m < 4; ++m) { acc[m][0] = vzero; acc[m][1] = vzero; }

    // A (16x4 f32): lanes 0-15 M=0..15 K={4k,4k+1}; lanes 16-31 K={4k+2,4k+3}.
    // B (4x16 f32): lanes 0-15 K=4k+2hi N=lane; B[k][n] = W_e[n][k].
    const float* aptr  = &Xs[lo * XS_STRIDE + 2 * hi];
    const float* bptrA = &Ws[(n0a + lo) * XS_STRIDE + 2 * hi];
    const float* bptrB = &Ws[(n0b + lo) * XS_STRIDE + 2 * hi];

#pragma unroll 4
    for (int k = 0; k < D_ / 4; ++k) {
        v2f b0 = *(const v2f*)(bptrA + 4 * k);
        v2f b1 = *(const v2f*)(bptrB + 4 * k);
#pragma unroll
        for (int m = 0; m < 4; ++m) {
            v2f av = *(const v2f*)(aptr + m * 16 * XS_STRIDE + 4 * k);
            acc[m][0] = __builtin_amdgcn_wmma_f32_16x16x4_f32(
                false, av, false, b0, (short)0, acc[m][0], false, false);
            acc[m][1] = __builtin_amdgcn_wmma_f32_16x16x4_f32(
                false, av, false, b1, (short)0, acc[m][1], false, false);
        }
    }

    // Epilogue: C/D layout — VGPR g: lanes0-15 M=g, lanes16-31 M=g+8; N=lane&15.
    float biasA = Bs[n0a + lo];
    float biasB = Bs[n0b + lo];
#pragma unroll
    for (int m = 0; m < 4; ++m) {
#pragma unroll
        for (int g = 0; g < 8; ++g) {
            int row = m * 16 + g + hi * 8;
            int tk = Toks[row];
            if (tk >= 0) {
                float w = Wts[row];
                atomicAdd(&out[tk * D_ + n0a + lo], (acc[m][0][g] + biasA) * w);
                if (has2)
                    atomicAdd(&out[tk * D_ + n0b + lo], (acc[m][1][g] + biasB) * w);
            }
        }
    }
}

// ---------------------------------------------------------------------------
extern "C" void kernel_launch(void* const* d_in, const int* in_sizes, int n_in,
                              void* d_out, int out_size, void* d_ws, size_t ws_size,
                              hipStream_t stream) {
    const float* x  = (const float*)d_in[0];   // [B,S,D]
    const float* gw = (const float*)d_in[1];   // [E,D]
    const float* gb = (const float*)d_in[2];   // [E]
    const float* ew = (const float*)d_in[3];   // [E,D,D]
    const float* eb = (const float*)d_in[4];   // [E,D]
    float* out = (float*)d_out;                // [B,S,D]

    // Workspace layout: counters | token lists | weight lists (~2.1 MB).
    char* ws = (char*)d_ws;
    int*   cnt = (int*)ws;                                        // E_ ints
    int*   tok = (int*)(ws + 256);                                // E_*CAP ints
    float* wt  = (float*)(ws + 256 + sizeof(int) * E_ * CAP);     // E_*CAP floats

    moe_zero_kernel<<<2048, 256, 0, stream>>>(out, NTOK * D_, cnt);
    moe_gate_kernel<<<NTOK / 8, 256, 0, stream>>>(x, gw, gb, cnt, tok, wt);

    const int smem_bytes =
        (TILE_TOK * XS_STRIDE + D_ * XS_STRIDE + D_ + TILE_TOK) * (int)sizeof(float)
        + TILE_TOK * (int)sizeof(int);  // 298,176 B < 320 KB WGP LDS

    dim3 grid2(CAP / TILE_TOK, E_);     // 512 x 8; inactive tiles exit early
    moe_expert_kernel<<<grid2, BLOCK2, smem_bytes, stream>>>(
        x, ew, eb, cnt, tok, wt, out);
}